// VRCA_20469814133195
// MI455X (gfx1250) — compile-verified
//
#include <hip/hip_runtime.h>
#include <hip/hip_bf16.h>
#include <math.h>

typedef __attribute__((ext_vector_type(2))) float v2f;
typedef __attribute__((ext_vector_type(8))) float v8f;

#define NB 4      // batch
#define NSEQ 512  // sequence length
#define ND 64     // model dim

// ---------------------------------------------------------------------------
// GEMM: C[M,N] = act(A[M,K] @ W[N,K]^T + bias[N]) using V_WMMA_F32_16X16X4_F32.
// One wave per 16x16 output tile; K stepped by 4.
// A frag layout (16x4 f32): lane M = lane&15, VGPR v holds K = v + 2*(lane>>4).
// B frag layout (4x16 f32): lane N = lane&15, VGPR v holds K = v + 2*(lane>>4).
// C/D (16x16 f32): VGPR r, lane l -> M = r + 8*(l>>4), N = l&15.
// ---------------------------------------------------------------------------
__global__ void cdna5_proj_kernel(const float* __restrict__ A,
                                  const float* __restrict__ W,
                                  const float* __restrict__ bias,
                                  float* __restrict__ C,
                                  int M, int N, int K, int act)
{
    const int lane   = threadIdx.x & 31;
    const int wave   = blockIdx.x * (blockDim.x >> 5) + (threadIdx.x >> 5);
    const int ntiles = N >> 4;
    const int m0     = (wave / ntiles) << 4;
    const int n0     = (wave % ntiles) << 4;
    if (m0 >= M) return;
    const int mr = m0 + (lane & 15);
    const int nc = n0 + (lane & 15);
    const int kh = (lane >> 4) << 1;

    v8f acc = {};
    for (int k0 = 0; k0 < K; k0 += 4) {
        v2f a, b;
        a.x = A[mr * K + k0 + kh];
        a.y = A[mr * K + k0 + kh + 1];
        b.x = W[nc * K + k0 + kh];        // B = W^T : B[k][n] = W[n][k]
        b.y = W[nc * K + k0 + kh + 1];
        acc = __builtin_amdgcn_wmma_f32_16x16x4_f32(false, a, false, b,
                                                    (short)0, acc, false, false);
    }
    const float bn = bias[nc];
    for (int r = 0; r < 8; ++r) {
        float v = acc[r] + bn;
        if (act == 1) v = 0.5f * v * (1.0f + erff(v * 0.70710678118654752f)); // exact GELU
        C[(m0 + r + ((lane >> 4) << 3)) * N + nc] = v;
    }
}

// ---------------------------------------------------------------------------
// Row softmax over last dim (64): one wave32 per row, 2 elements per lane.
// ---------------------------------------------------------------------------
__global__ void softmax64_kernel(float* __restrict__ X)
{
    const int lane = threadIdx.x & 31;
    const int row  = blockIdx.x * (blockDim.x >> 5) + (threadIdx.x >> 5);
    float* p = X + row * 64;
    float a = p[lane], b = p[lane + 32];
    float m = fmaxf(a, b);
    for (int o = 16; o > 0; o >>= 1) m = fmaxf(m, __shfl_xor(m, o, 32));
    float ea = expf(a - m), eb = expf(b - m);
    float s = ea + eb;
    for (int o = 16; o > 0; o >>= 1) s += __shfl_xor(s, o, 32);
    s = 1.0f / s;
    p[lane]      = ea * s;
    p[lane + 32] = eb * s;
}

// ---------------------------------------------------------------------------
// Stable rank-sort of 64-wide rows: thread a computes the stable rank of its
// element (ties broken by original index, matching jnp.argsort) and scatters
// (value, original index) into sorted position. 4 rows per 256-thread block.
// ---------------------------------------------------------------------------
__global__ void ranksort64_kernel(const float* __restrict__ X,
                                  float* __restrict__ SV, int* __restrict__ SI)
{
    __shared__ float sm[4][64];
    const int rl  = threadIdx.x >> 6;
    const int a   = threadIdx.x & 63;
    const int row = (blockIdx.x << 2) + rl;
    sm[rl][a] = X[row * 64 + a];
    __syncthreads();
    const float va = sm[rl][a];
    int rank = 0;
    for (int c = 0; c < 64; ++c) {
        float vc = sm[rl][c];
        rank += (int)((vc < va) || (vc == va && c < a));
    }
    SV[row * 64 + rank] = va;
    SI[row * 64 + rank] = a;
}

// ---------------------------------------------------------------------------
// Cucconi scores + row softmax + attn = S @ V, fused.
// Block = (batch b, 16-row i-tile), 256 threads (8 waves).
// Phase 1: stage sorted k rows (16) in LDS.
// Phase 2: per 64-column q chunk staged in LDS, each pair (i,j) does a stable
//          two-pointer merge of the two sorted 64-lists, accumulating
//          sum((idx+1)^2) / sum((129-idx-1)^2) over the first 64 merged
//          elements == the 64 smallest of concat(q[j],k[i]).
// Phase 3: softmax(-log(score+eps)) per row over 512 (wave32 reductions).
// Phase 4: attn tile (16x64) = S(16x512) @ V(512x64) via WMMA f32 16x16x4,
//          A read from LDS, one wave per 16-col N tile.
// ---------------------------------------------------------------------------
__global__ void cucconi_attn_kernel(const float* __restrict__ qsv,
                                    const int*   __restrict__ qsi,
                                    const float* __restrict__ ksv,
                                    const int*   __restrict__ ksi,
                                    const float* __restrict__ V,
                                    float* __restrict__ attn,
                                    float mu, float inv_sigma, float rio,
                                    float half_inv, float sup)
{
    __shared__ float         S[16][512];   // 32 KB score tile
    __shared__ float         kvs[16][64];  // sorted k values
    __shared__ unsigned char kis[16][64];  // sorted k original indices (0..63)
    __shared__ float         qvs[64][64];  // q chunk sorted values
    __shared__ unsigned char qis[64][64];  // q chunk original indices

    const int b   = blockIdx.x >> 5;       // 32 i-tiles per batch
    const int i0  = (blockIdx.x & 31) << 4;
    const int tid = threadIdx.x;

    for (int t = tid; t < 16 * 64; t += 256) {
        int r = t >> 6, c = t & 63;
        int g = (b * NSEQ + i0 + r) * 64 + c;
        kvs[r][c] = ksv[g];
        kis[r][c] = (unsigned char)ksi[g];
    }
    __syncthreads();

    for (int ch = 0; ch < 8; ++ch) {
        const int j0 = ch << 6;
        for (int t = tid; t < 64 * 64; t += 256) {
            int r = t >> 6, c = t & 63;
            int g = (b * NSEQ + j0 + r) * 64 + c;
            qvs[r][c] = qsv[g];
            qis[r][c] = (unsigned char)qsi[g];
        }
        __syncthreads();

        for (int p = 0; p < 4; ++p) {
            const int pair = tid + (p << 8);   // 0..1023
            const int r  = pair >> 6;          // i-row 0..15
            const int jj = pair & 63;          // j within chunk
            const float*         qv = qvs[jj];
            const unsigned char* qi = qis[jj];
            const float*         kv = kvs[r];
            const unsigned char* ki = kis[r];
            int pq = 0, pk = 0;
            float ss = 0.f, rss = 0.f;
            for (int t2 = 0; t2 < 64; ++t2) {
                float qval = (pq < 64) ? qv[pq] : 3.0e38f;
                float kval = (pk < 64) ? kv[pk] : 3.0e38f;
                bool takeq = (qval <= kval);   // tie -> q (lower original index)
                int  pidx  = takeq ? ((int)qi[pq] + 1) : ((int)ki[pk] + 65);
                pq += takeq ? 1 : 0;
                pk += takeq ? 0 : 1;
                float fp = (float)pidx;        // <=128; sums < 2^24, exact in f32
                float rp = (float)(129 - pidx);
                ss  += fp * fp;
                rss += rp * rp;
            }
            float u = (6.f * ss  - mu) * inv_sigma;
            float v = (6.f * rss - mu) * inv_sigma;
            float c = (u * u + v * v - 2.f * rio * u * v) * half_inv;
            c = (c > sup) ? sup : c;
            S[r][j0 + jj] = -logf(c + 1e-6f);
        }
        __syncthreads();
    }

    // Phase 3: softmax rows (8 waves x 2 rows)
    const int wid = tid >> 5, lane = tid & 31;
    for (int rr = 0; rr < 2; ++rr) {
        const int r = wid * 2 + rr;
        float m = -3.0e38f;
        for (int t = lane; t < 512; t += 32) m = fmaxf(m, S[r][t]);
        for (int o = 16; o > 0; o >>= 1) m = fmaxf(m, __shfl_xor(m, o, 32));
        float s = 0.f;
        for (int t = lane; t < 512; t += 32) {
            float e = expf(S[r][t] - m);
            S[r][t] = e;
            s += e;
        }
        for (int o = 16; o > 0; o >>= 1) s += __shfl_xor(s, o, 32);
        s = 1.0f / s;
        for (int t = lane; t < 512; t += 32) S[r][t] *= s;
    }
    __syncthreads();

    // Phase 4: attn(16x64) = S(16x512) @ V[b](512x64); one wave per N tile.
    if (wid < 4) {
        const int n0 = wid << 4;
        const int nc = n0 + (lane & 15);
        const int kh = (lane >> 4) << 1;
        const int ar = lane & 15;
        const float* Vb = V + b * NSEQ * 64;
        v8f acc = {};
        for (int k0 = 0; k0 < 512; k0 += 4) {
            v2f a, bb;
            a.x  = S[ar][k0 + kh];
            a.y  = S[ar][k0 + kh + 1];
            bb.x = Vb[(k0 + kh) * 64 + nc];
            bb.y = Vb[(k0 + kh + 1) * 64 + nc];
            acc = __builtin_amdgcn_wmma_f32_16x16x4_f32(false, a, false, bb,
                                                        (short)0, acc, false, false);
        }
        for (int r = 0; r < 8; ++r) {
            int row = i0 + r + ((lane >> 4) << 3);
            attn[(b * NSEQ + row) * 64 + nc] = acc[r];
        }
    }
}

// ---------------------------------------------------------------------------
// out = LayerNorm(x + r) * g + b over last dim 64; one wave32 per row.
// ---------------------------------------------------------------------------
__global__ void add_ln_kernel(const float* __restrict__ X,
                              const float* __restrict__ R,
                              const float* __restrict__ g,
                              const float* __restrict__ bvec,
                              float* __restrict__ O)
{
    const int lane = threadIdx.x & 31;
    const int row  = blockIdx.x * (blockDim.x >> 5) + (threadIdx.x >> 5);
    const int base = row * 64;
    float a0 = X[base + lane]      + R[base + lane];
    float a1 = X[base + lane + 32] + R[base + lane + 32];
    float s = a0 + a1;
    for (int o = 16; o > 0; o >>= 1) s += __shfl_xor(s, o, 32);
    const float mean = s * (1.0f / 64.0f);
    float d0 = a0 - mean, d1 = a1 - mean;
    float vs = d0 * d0 + d1 * d1;
    for (int o = 16; o > 0; o >>= 1) vs += __shfl_xor(vs, o, 32);
    const float inv = rsqrtf(vs * (1.0f / 64.0f) + 1e-5f);
    O[base + lane]      = d0 * inv * g[lane]      + bvec[lane];
    O[base + lane + 32] = d1 * inv * g[lane + 32] + bvec[lane + 32];
}

// ---------------------------------------------------------------------------
extern "C" void kernel_launch(void* const* d_in, const int* in_sizes, int n_in,
                              void* d_out, int out_size, void* d_ws, size_t ws_size,
                              hipStream_t stream)
{
    (void)in_sizes; (void)n_in; (void)out_size; (void)ws_size;
    const float* x    = (const float*)d_in[0];
    const float* Wq   = (const float*)d_in[1];
    const float* bq   = (const float*)d_in[2];
    const float* Wk   = (const float*)d_in[3];
    const float* bk   = (const float*)d_in[4];
    const float* Wv   = (const float*)d_in[5];
    const float* bv   = (const float*)d_in[6];
    const float* Wo   = (const float*)d_in[7];
    const float* bo   = (const float*)d_in[8];
    const float* W1   = (const float*)d_in[9];
    const float* b1   = (const float*)d_in[10];
    const float* W2   = (const float*)d_in[11];
    const float* b2   = (const float*)d_in[12];
    const float* ln1g = (const float*)d_in[13];
    const float* ln1b = (const float*)d_in[14];
    const float* ln2g = (const float*)d_in[15];
    const float* ln2b = (const float*)d_in[16];
    float* out = (float*)d_out;

    // Workspace layout (elements); total 8 MB.
    float* ws   = (float*)d_ws;
    float* q    = ws;
    float* k    = ws + 131072;
    float* vt   = ws + 262144;
    float* v2   = ws + 393216;
    float* qsv  = ws + 524288;
    float* ksv  = ws + 655360;
    float* attn = ws + 786432;
    float* nx   = ws + 917504;
    float* x1   = ws + 1048576;
    float* y2   = ws + 1179648;
    float* h1   = ws + 1310720;                 // 2048x256
    int*   qsi  = (int*)(ws + 1835008);
    int*   ksi  = (int*)(ws + 1966080);

    // Cucconi constants (double on host, passed as f32).
    const double Dd    = 64.0;
    const double rio   = 2.0 * (4.0 * Dd * Dd - 4.0) / ((4.0 * Dd + 1.0) * (16.0 * Dd + 11.0)) - 1.0;
    const double sup   = (-8.0 * log(0.05)) / 4.0 / (1.0 - rio);
    const double mu    = Dd * (2.0 * Dd + 1.0) * (4.0 * Dd + 1.0);
    const double sigma = sqrt(Dd * Dd * (2.0 * Dd + 1.0) * (4.0 * Dd + 1.0) * (16.0 * Dd + 11.0) / 5.0);
    const float  f_mu   = (float)mu;
    const float  f_isig = (float)(1.0 / sigma);
    const float  f_rio  = (float)rio;
    const float  f_hin  = (float)(0.5 / (1.0 - rio * rio));
    const float  f_sup  = (float)sup;

    const int M = NB * NSEQ;   // 2048
    dim3 blk(256);

    // Projections (x @ W^T + b) — v projected twice per reference.
    cdna5_proj_kernel<<<64,  blk, 0, stream>>>(x,  Wq, bq, q,  M, 64, 64, 0);
    cdna5_proj_kernel<<<64,  blk, 0, stream>>>(x,  Wk, bk, k,  M, 64, 64, 0);
    cdna5_proj_kernel<<<64,  blk, 0, stream>>>(x,  Wv, bv, vt, M, 64, 64, 0);
    cdna5_proj_kernel<<<64,  blk, 0, stream>>>(vt, Wv, bv, v2, M, 64, 64, 0);

    softmax64_kernel<<<256, blk, 0, stream>>>(q);
    softmax64_kernel<<<256, blk, 0, stream>>>(k);

    ranksort64_kernel<<<512, blk, 0, stream>>>(q, qsv, qsi);
    ranksort64_kernel<<<512, blk, 0, stream>>>(k, ksv, ksi);

    cucconi_attn_kernel<<<NB * (NSEQ / 16), blk, 0, stream>>>(
        qsv, qsi, ksv, ksi, v2, attn, f_mu, f_isig, f_rio, f_hin, f_sup);

    cdna5_proj_kernel<<<64,  blk, 0, stream>>>(attn, Wo, bo, nx, M, 64, 64, 0);
    add_ln_kernel<<<256, blk, 0, stream>>>(x, nx, ln1g, ln1b, x1);

    cdna5_proj_kernel<<<256, blk, 0, stream>>>(x1, W1, b1, h1, M, 256, 64, 1);  // + exact GELU
    cdna5_proj_kernel<<<64,  blk, 0, stream>>>(h1, W2, b2, y2, M, 64, 256, 0);
    add_ln_kernel<<<256, blk, 0, stream>>>(x1, y2, ln2g, ln2b, out);
}